// GridSet_86818468921553
// MI455X (gfx1250) — compile-verified
//
#include <hip/hip_runtime.h>

// ---------------------------------------------------------------------------
// Coiflet-4 synthesis filters (L = 24), zero-padded to 96 entries with the
// 24 real taps at offset 40. Index  g[t]  ->  GxP[t + 40]; every index used
// by the kernels (t in [-39, 38]) lands in [1, 78]: no bounds check, no
// exec-mask branching when building WMMA filter operands.
// G0[k] = COIF4_DEC_LO[23-k]   (reversed lowpass)
// G1[k] = COIF4_DEC_LO[k] * (-1)^k  (highpass)
// ---------------------------------------------------------------------------
#define ZPAD40 0.f,0.f,0.f,0.f,0.f,0.f,0.f,0.f,0.f,0.f, \
               0.f,0.f,0.f,0.f,0.f,0.f,0.f,0.f,0.f,0.f, \
               0.f,0.f,0.f,0.f,0.f,0.f,0.f,0.f,0.f,0.f, \
               0.f,0.f,0.f,0.f,0.f,0.f,0.f,0.f,0.f,0.f
static __device__ __constant__ float G0P[96] = {
    ZPAD40,
     0.0008923136685823146f,   -0.0016294920126017326f,  -0.0073461663276420935f,
     0.016068943964776348f,     0.026682300156053072f,   -0.08126669968087875f,
    -0.05607731331675481f,      0.41530840703043026f,     0.782238930920499f,
     0.4343860564914685f,      -0.06662747426342504f,    -0.09622044203398798f,
     0.03933442712333749f,      0.025082261844864097f,   -0.015211731527946259f,
    -0.00565828668661072f,      0.003751436157278457f,    0.0012665619292989445f,
    -0.0005890207562443383f,   -0.00025997455248771324f,  6.233903446100713e-05f,
     3.1229875865345646e-05f,  -3.2596802368833675e-06f, -1.7849850030882614e-06f
};
static __device__ __constant__ float G1P[96] = {
    ZPAD40,
    -1.7849850030882614e-06f,   3.2596802368833675e-06f,  3.1229875865345646e-05f,
    -6.233903446100713e-05f,   -0.00025997455248771324f,  0.0005890207562443383f,
     0.0012665619292989445f,   -0.003751436157278457f,   -0.00565828668661072f,
     0.015211731527946259f,     0.025082261844864097f,   -0.03933442712333749f,
    -0.09622044203398798f,      0.06662747426342504f,     0.4343860564914685f,
    -0.782238930920499f,        0.41530840703043026f,     0.05607731331675481f,
    -0.08126669968087875f,     -0.026682300156053072f,    0.016068943964776348f,
     0.0073461663276420935f,   -0.0016294920126017326f,  -0.0008923136685823146f
};
#undef ZPAD40

typedef __attribute__((ext_vector_type(2))) float v2f;
typedef __attribute__((ext_vector_type(4))) float v4f;
typedef __attribute__((ext_vector_type(8))) float v8f;

#define WMMA_F32(A, B, C) \
    __builtin_amdgcn_wmma_f32_16x16x4_f32(false, (A), false, (B), (short)0, (C), false, false)

// ---------------------------------------------------------------------------
// Pass 1: column synthesis as fp32 WMMA, filter on the A side.
//   out16x16 = Filt(16x32) x Data(32x16)
//   Filt[r][k] = g[r + 23 - 2k]   (tile-invariant; band scale folded in)
//   Data[k][x] = band[(ibase+k) & (n-1)][x0+x],  ibase = 8*ty - 6
// Result pair (loP, hiP) is stored INTERLEAVED: lohi[y][2x]=lo, [2x+1]=hi
// (8 coalesced b64 stores). n is a template constant so all loads/stores use
// immediate instruction offsets instead of 64-bit VALU address chains.
// yh layout: (C, 3, n, n) -> lh = band 0, hl = band 1, hh = band 2.
// ---------------------------------------------------------------------------
template<int n>
__global__ __launch_bounds__(128) void colsyn_wmma_kernel(
    const float* __restrict__ ll, const float* __restrict__ yh,
    float* __restrict__ lohi, float hscale)
{
    constexpr int    N    = 2 * n;
    constexpr int    mask = n - 1;
    constexpr size_t nn   = (size_t)n * n;

    const int lane = threadIdx.x;
    const int half = lane >> 4;
    const int l15  = lane & 15;
    const int tx   = blockIdx.x;                        // x tile   [0, n/16)
    const int ty   = blockIdx.y * 4 + threadIdx.y;      // y tile   [0, N/16)
    const int c    = blockIdx.z;

    const float* __restrict__ llc = ll + (size_t)c * nn;
    const float* __restrict__ lhc = yh + ((size_t)c * 3 + 0) * nn;
    const float* __restrict__ hlc = yh + ((size_t)c * 3 + 1) * nn;
    const float* __restrict__ hhc = yh + ((size_t)c * 3 + 2) * nn;

    const int xcol   = tx * 16 + l15;
    const int ibase  = 8 * ty - 6;
    const int k0base = 2 * half;

    // Filter A-operands (16x4 layout: lane<16 -> row=l15,K={0,1};
    // lane>=16 -> row=l15,K={2,3}); band scale folded in.
    v2f aG0[8], aG0s[8], aG1s[8];
#pragma unroll
    for (int kc = 0; kc < 8; ++kc) {
        const int idx = l15 + 63 - 2 * (4 * kc + k0base);   // 63 = 23 + 40 pad
        const float g0x = G0P[idx], g0y = G0P[idx - 2];
        const float g1x = G1P[idx], g1y = G1P[idx - 2];
        aG0[kc].x  = g0x;            aG0[kc].y  = g0y;
        aG0s[kc].x = hscale * g0x;   aG0s[kc].y = hscale * g0y;
        aG1s[kc].x = hscale * g1x;   aG1s[kc].y = hscale * g1y;
    }

    v8f accLo = {0.f,0.f,0.f,0.f,0.f,0.f,0.f,0.f};
    v8f accHi = {0.f,0.f,0.f,0.f,0.f,0.f,0.f,0.f};

    if (ibase >= 0 && ibase + 32 <= n) {           // wave-uniform fast path
        const float* pLL = llc + (size_t)(ibase + k0base) * n + xcol;
        const float* pLH = lhc + (size_t)(ibase + k0base) * n + xcol;
        const float* pHL = hlc + (size_t)(ibase + k0base) * n + xcol;
        const float* pHH = hhc + (size_t)(ibase + k0base) * n + xcol;
#pragma unroll
        for (int kc = 0; kc < 8; ++kc) {
            constexpr int s = 1;  (void)s;
            const int o0 = 4 * kc * n;             // compile-time constants ->
            const int o1 = o0 + n;                 // immediate load offsets
            v2f bLL, bLH, bHL, bHH;
            bLL.x = pLL[o0];  bLL.y = pLL[o1];
            bLH.x = pLH[o0];  bLH.y = pLH[o1];
            bHL.x = pHL[o0];  bHL.y = pHL[o1];
            bHH.x = pHH[o0];  bHH.y = pHH[o1];
            accLo = WMMA_F32(aG0[kc],  bLL, accLo);
            accLo = WMMA_F32(aG1s[kc], bLH, accLo);
            accHi = WMMA_F32(aG0s[kc], bHL, accHi);
            accHi = WMMA_F32(aG1s[kc], bHH, accHi);
        }
    } else {                                       // periodic-wrap boundary
#pragma unroll
        for (int kc = 0; kc < 8; ++kc) {
            const int k0 = 4 * kc + k0base;
            const size_t o0 = (size_t)((ibase + k0)     & mask) * n + xcol;
            const size_t o1 = (size_t)((ibase + k0 + 1) & mask) * n + xcol;
            v2f bLL, bLH, bHL, bHH;
            bLL.x = llc[o0];  bLL.y = llc[o1];
            bLH.x = lhc[o0];  bLH.y = lhc[o1];
            bHL.x = hlc[o0];  bHL.y = hlc[o1];
            bHH.x = hhc[o0];  bHH.y = hhc[o1];
            accLo = WMMA_F32(aG0[kc],  bLL, accLo);
            accLo = WMMA_F32(aG1s[kc], bLH, accLo);
            accHi = WMMA_F32(aG0s[kc], bHL, accHi);
            accHi = WMMA_F32(aG1s[kc], bHH, accHi);
        }
    }

    // Interleaved store: acc[v] -> row 16*ty + v + 8*half, pair at 2*xcol.
    float* __restrict__ outRow =
        lohi + ((size_t)c * N + (size_t)(16 * ty + 8 * half)) * (2 * n) + 2 * xcol;
#pragma unroll
    for (int v = 0; v < 8; ++v) {
        v2f pr;  pr.x = accLo[v];  pr.y = accHi[v];
        *(v2f*)(outRow + (size_t)v * (2 * n)) = pr;   // aligned b64
    }
}

// ---------------------------------------------------------------------------
// Pass 2: row synthesis as fp32 WMMA, filter on the B side.
//   out16x16 = Data(16x32) x Filt(32x16),  Filt[k][j] = g[j + 23 - 2k]
// Interleaved lohi input: one aligned b128 load per chunk yields
// {lo[i], hi[i], lo[i+1], hi[i+1]} = both A operands. Only the 4 boundary
// column-tiles take the masked periodic-wrap path (2 b64 loads per chunk).
// ---------------------------------------------------------------------------
template<int n>
__global__ __launch_bounds__(128) void rowsyn_wmma_kernel(
    const float* __restrict__ lohi, float* __restrict__ outp)
{
    constexpr int N    = 2 * n;
    constexpr int mask = n - 1;

    const int lane = threadIdx.x;
    const int half = lane >> 4;
    const int l15  = lane & 15;
    const int tp   = blockIdx.x;                    // column tile [0, N/16)
    const int tr   = blockIdx.y * 4 + threadIdx.y;  // row tile    [0, N/16)
    const int c    = blockIdx.z;

    const float* __restrict__ src = lohi + (size_t)c * N * (2 * n);
    float* __restrict__       out = outp + (size_t)c * N * N;

    const int row0   = tr * 16;
    const int ibase  = 8 * tp - 6;
    const int k0base = 2 * half;

    // Filter B-operands (4x16 layout: b.x -> row 2*half, b.y -> row 2*half+1,
    // column = l15), built branch-free from the padded tables.
    v2f b0[8], b1[8];
#pragma unroll
    for (int kc = 0; kc < 8; ++kc) {
        const int idx = l15 + 63 - 2 * (4 * kc + k0base);
        b0[kc].x = G0P[idx];  b0[kc].y = G0P[idx - 2];
        b1[kc].x = G1P[idx];  b1[kc].y = G1P[idx - 2];
    }

    const float* __restrict__ rowBase = src + (size_t)(row0 + l15) * (2 * n);

    v8f acc = {0.f,0.f,0.f,0.f,0.f,0.f,0.f,0.f};

    if (ibase >= 0 && ibase + 32 <= n) {           // wave-uniform fast path
        // element i = ibase + k0base; 16-byte aligned (2*(even) floats)
        const float* p = rowBase + 2 * (ibase + k0base);
#pragma unroll
        for (int kc = 0; kc < 8; ++kc) {
            const v4f q = *(const v4f*)(p + 8 * kc);   // immediate offset b128
            v2f aLo, aHi;
            aLo.x = q.x;  aLo.y = q.z;                 // lo[i], lo[i+1]
            aHi.x = q.y;  aHi.y = q.w;                 // hi[i], hi[i+1]
            acc = WMMA_F32(aLo, b0[kc], acc);
            acc = WMMA_F32(aHi, b1[kc], acc);
        }
    } else {                                       // periodic-wrap boundary
#pragma unroll
        for (int kc = 0; kc < 8; ++kc) {
            const int k0 = 4 * kc + k0base;
            const int i0 = (ibase + k0)     & mask;
            const int i1 = (ibase + k0 + 1) & mask;
            const v2f p0 = *(const v2f*)(rowBase + 2 * i0);  // {lo,hi}[i0]
            const v2f p1 = *(const v2f*)(rowBase + 2 * i1);  // {lo,hi}[i1]
            v2f aLo, aHi;
            aLo.x = p0.x;  aLo.y = p1.x;
            aHi.x = p0.y;  aHi.y = p1.y;
            acc = WMMA_F32(aLo, b0[kc], acc);
            acc = WMMA_F32(aHi, b1[kc], acc);
        }
    }

    float* __restrict__ outBase = out + (size_t)(row0 + 8 * half) * N + 16 * tp + l15;
#pragma unroll
    for (int v = 0; v < 8; ++v) {
        outBase[(size_t)v * N] = acc[v];           // immediate offset stores
    }
}

// ---------------------------------------------------------------------------
// Pass 3: bilinear grid sample. One wave per point, lane = channel
// (C == 32 == wave32): the 32 output floats per point form one coalesced
// 128-byte store.
// ---------------------------------------------------------------------------
__global__ __launch_bounds__(256) void sample_kernel(
    const float* __restrict__ pts, const float* __restrict__ plane,
    float* __restrict__ out, int n_pts)
{
    const int tid = blockIdx.x * 256 + threadIdx.x;
    const int p   = tid >> 5;
    const int c   = tid & 31;
    if (p >= n_pts) return;

    const int W = 1024, H = 1024;
    const float px = pts[2 * p + 0];
    const float py = pts[2 * p + 1];
    float x = fminf(fmaxf((px + 1.0f) * 0.5f * (float)(W - 1), 0.0f), (float)(W - 1));
    float y = fminf(fmaxf((py + 1.0f) * 0.5f * (float)(H - 1), 0.0f), (float)(H - 1));
    const int x0 = (int)floorf(x);
    const int y0 = (int)floorf(y);
    const int x1 = min(x0 + 1, W - 1);
    const int y1 = min(y0 + 1, H - 1);
    const float wx = x - (float)x0;
    const float wy = y - (float)y0;

    const float* __restrict__ pc = plane + (size_t)c * W * H;
    const float v00 = pc[(size_t)y0 * W + x0];
    const float v01 = pc[(size_t)y0 * W + x1];
    const float v10 = pc[(size_t)y1 * W + x0];
    const float v11 = pc[(size_t)y1 * W + x1];

    const float r = v00 * (1.f - wy) * (1.f - wx)
                  + v01 * (1.f - wy) * wx
                  + v10 * wy * (1.f - wx)
                  + v11 * wy * wx;
    out[(size_t)p * 32 + c] = r;
}

// ---------------------------------------------------------------------------
// Per-level launcher (n is compile-time).
// ---------------------------------------------------------------------------
template<int n>
static inline void run_level(const float* curLL, const float* yh,
                             float* lohiBuf, float* planeBuf,
                             float hscale, hipStream_t stream)
{
    constexpr int N = 2 * n;
    dim3 gc(n / 16, N / 64, 32);
    colsyn_wmma_kernel<n><<<gc, dim3(32, 4), 0, stream>>>(curLL, yh, lohiBuf, hscale);
    dim3 gr(N / 16, N / 64, 32);
    rowsyn_wmma_kernel<n><<<gr, dim3(32, 4), 0, stream>>>(lohiBuf, planeBuf);
}

// ---------------------------------------------------------------------------
// Driver. Inputs (setup_inputs order): 0=pts, 1=yl, 2=yh0, 3=yh1, 4=yh2.
// Workspace layout (268 MiB):
//   planeBuf : 32*1024*1024 f32  (ll at every level, final plane)
//   lohiBuf  : 32*1024*1024 f32  (interleaved lo/hi column-pass temp)
// ---------------------------------------------------------------------------
extern "C" void kernel_launch(void* const* d_in, const int* in_sizes, int n_in,
                              void* d_out, int out_size, void* d_ws, size_t ws_size,
                              hipStream_t stream)
{
    (void)n_in; (void)out_size; (void)ws_size;
    const float* pts = (const float*)d_in[0];
    const float* yl  = (const float*)d_in[1];
    const float* yh0 = (const float*)d_in[2];
    const float* yh1 = (const float*)d_in[3];
    const float* yh2 = (const float*)d_in[4];

    float* planeBuf = (float*)d_ws;
    float* lohiBuf  = planeBuf + (size_t)32 * 1024 * 1024;

    run_level<128>(yl,       yh2, lohiBuf, planeBuf, 0.6f, stream);
    run_level<256>(planeBuf, yh1, lohiBuf, planeBuf, 0.4f, stream);
    run_level<512>(planeBuf, yh0, lohiBuf, planeBuf, 0.2f, stream);

    const int n_pts = in_sizes[0] / 2;
    const int total = n_pts * 32;
    sample_kernel<<<(total + 255) / 256, 256, 0, stream>>>(
        pts, planeBuf, (float*)d_out, n_pts);
}